// Layer6bTransformer_49761491091509
// MI455X (gfx1250) — compile-verified
//
#include <hip/hip_runtime.h>
#include <hip/hip_bf16.h>
#include <math.h>

#define SDR_DIM 65536
#define HIDDEN  4096
#define N_MOD   2048
#define PHASE   4096
#define NORM_EPS 1e-8f

typedef float v2f __attribute__((ext_vector_type(2)));
typedef float v4f __attribute__((ext_vector_type(4)));
typedef float v8f __attribute__((ext_vector_type(8)));

// ---------------------------------------------------------------------------
// Stage 1: h = relu(W1 @ x + b1).  W1 is 4096 x 65536 f32 (1 GB) -> pure HBM
// streaming.  One 256-thread block per row, coalesced 16B non-temporal loads,
// LDS tree reduction.  x (256 KB) is L2-resident and reused by every block.
// ---------------------------------------------------------------------------
__global__ __launch_bounds__(256)
void gemv_relu_kernel(const float* __restrict__ W1, const float* __restrict__ x,
                      const float* __restrict__ b1, float* __restrict__ h) {
  __shared__ float red[256];
  const int row = blockIdx.x;
  const int t   = threadIdx.x;
  const v4f* __restrict__ Wr = (const v4f*)(W1 + (size_t)row * SDR_DIM);
  const v4f* __restrict__ xv = (const v4f*)x;

  float acc = 0.0f;
  const int n4 = SDR_DIM / 4;            // 16384 float4s per row
#pragma unroll 4
  for (int i = t; i < n4; i += 256) {
    // Prefetch next block-wide tile of the streamed matrix.
    if (i + 256 < n4) __builtin_prefetch(&Wr[i + 256], 0, 0);
    v4f w = __builtin_nontemporal_load(&Wr[i]);   // stream W1, don't pollute L2
    v4f v = xv[i];                                 // cached: reused by all rows
    acc = fmaf(w.x, v.x, acc);
    acc = fmaf(w.y, v.y, acc);
    acc = fmaf(w.z, v.z, acc);
    acc = fmaf(w.w, v.w, acc);
  }

  red[t] = acc;
  __syncthreads();
  for (int s = 128; s > 0; s >>= 1) {
    if (t < s) red[t] += red[t + s];
    __syncthreads();
  }
  if (t == 0) h[row] = fmaxf(red[0] + b1[row], 0.0f);
}

// ---------------------------------------------------------------------------
// WMMA GEMV core (16 rows per wave, K stepped by 4 per v_wmma_f32_16x16x4_f32).
// A (16x4 f32, documented layout): lane = M, lane-half selects K {0,1}/{2,3},
// VGPR index enumerates K within the half -> one v2f load per lane per step.
// B (4x16): the input vector broadcast to all 16 columns -> same v2f in every
// lane of a half.  Then every D column holds the same 16 dot products; with
// the documented C/D layout lane 0 holds rows 0..7 (VGPR 0..7) and lane 16
// holds rows 8..15.  Two accumulator chains give ILP on the WMMA pipe.
// ---------------------------------------------------------------------------
__device__ __forceinline__
void wmma_gemv16(const float* __restrict__ W, const float* __restrict__ x,
                 int row0, int m, int half, int K, v8f& acc0, v8f& acc1) {
  const float* __restrict__ Arow = W + (size_t)(row0 + m) * K + 2 * half;
  const float* __restrict__ Bv   = x + 2 * half;
  for (int k = 0; k < K; k += 8) {
    v2f a0 = *(const v2f*)(Arow + k);
    v2f b0 = *(const v2f*)(Bv + k);
    v2f a1 = *(const v2f*)(Arow + k + 4);
    v2f b1 = *(const v2f*)(Bv + k + 4);
    acc0 = __builtin_amdgcn_wmma_f32_16x16x4_f32(false, a0, false, b0,
                                                 (short)0, acc0, false, false);
    acc1 = __builtin_amdgcn_wmma_f32_16x16x4_f32(false, a1, false, b1,
                                                 (short)0, acc1, false, false);
  }
}

// Stage 2: z_ego = W2 @ h + b2   (4096 x 4096)
__global__ __launch_bounds__(256)
void gemv_wmma_bias_kernel(const float* __restrict__ W, const float* __restrict__ x,
                           const float* __restrict__ bias, float* __restrict__ out,
                           int K) {
  const int wave = (blockIdx.x * blockDim.x + threadIdx.x) >> 5;
  const int lane = threadIdx.x & 31;
  const int m = lane & 15, half = lane >> 4;
  const int row0 = wave * 16;

  v8f acc0 = {0.f,0.f,0.f,0.f,0.f,0.f,0.f,0.f};
  v8f acc1 = {0.f,0.f,0.f,0.f,0.f,0.f,0.f,0.f};
  wmma_gemv16(W, x, row0, m, half, K, acc0, acc1);

  if (m == 0) {                       // lanes 0 and 16 hold the 16 row results
    const int rbase = row0 + 8 * half;
#pragma unroll
    for (int r = 0; r < 8; ++r)
      out[rbase + r] = acc0[r] + acc1[r] + bias[rbase + r];
  }
}

// Zero the two sum-of-squares accumulators.
__global__ void init_sums_kernel(float* __restrict__ sums) {
  if (threadIdx.x < 2) sums[threadIdx.x] = 0.0f;
}

// Stage 3: theta = vel_W @ velocity; rotate z_ego pairs -> z_flat; also
// accumulate ||z_ego||^2 into sums[0].
__global__ __launch_bounds__(256)
void rotate_kernel(const float* __restrict__ z_ego, const float* __restrict__ velocity,
                   const float* __restrict__ vel_W, float* __restrict__ z_flat,
                   float* __restrict__ sums) {
  __shared__ float red[256];
  const int t = threadIdx.x;
  const int i = blockIdx.x * 256 + t;          // 0 .. N_MOD-1
  const float v0 = velocity[0], v1 = velocity[1];

  float local = 0.0f;
  if (i < N_MOD) {
    float theta = fmaf(vel_W[2 * i], v0, vel_W[2 * i + 1] * v1);
    float c = cosf(theta), s = sinf(theta);
    float x = z_ego[2 * i], y = z_ego[2 * i + 1];
    z_flat[2 * i]     = fmaf(c, x, -s * y);
    z_flat[2 * i + 1] = fmaf(s, x,  c * y);
    local = fmaf(x, x, y * y);                 // target_norm uses z_ego
  }
  red[t] = local;
  __syncthreads();
  for (int s2 = 128; s2 > 0; s2 >>= 1) {
    if (t < s2) red[t] += red[t + s2];
    __syncthreads();
  }
  if (t == 0) atomicAdd(&sums[0], red[0]);
}

// Stage 4: h_new = tanh(W_tc@z_flat + W_ht@h_tc); z_fb = z_flat +
// 0.1*sigmoid(g)*h_new; accumulate ||z_fb||^2 into sums[1].
__global__ __launch_bounds__(256)
void gemv_wmma_fb_kernel(const float* __restrict__ W_tc, const float* __restrict__ z_flat,
                         const float* __restrict__ W_ht, const float* __restrict__ h_tc,
                         const float* __restrict__ g_nrt, float* __restrict__ z_fb,
                         float* __restrict__ sums, int K) {
  const int wave = (blockIdx.x * blockDim.x + threadIdx.x) >> 5;
  const int lane = threadIdx.x & 31;
  const int m = lane & 15, half = lane >> 4;
  const int row0 = wave * 16;

  v8f acc0 = {0.f,0.f,0.f,0.f,0.f,0.f,0.f,0.f};
  v8f acc1 = {0.f,0.f,0.f,0.f,0.f,0.f,0.f,0.f};
  wmma_gemv16(W_tc, z_flat, row0, m, half, K, acc0, acc1);  // W_tc @ z_flat
  wmma_gemv16(W_ht, h_tc,   row0, m, half, K, acc0, acc1);  // + W_ht @ h_tc

  if (m == 0) {
    const int rbase = row0 + 8 * half;
    float s = 0.0f;
#pragma unroll
    for (int r = 0; r < 8; ++r) {
      const int row = rbase + r;
      float hn  = tanhf(acc0[r] + acc1[r]);
      float sg  = 1.0f / (1.0f + expf(-g_nrt[row]));
      float zfb = fmaf(0.1f, sg * hn, z_flat[row]);
      z_fb[row] = zfb;
      s = fmaf(zfb, zfb, s);
    }
    atomicAdd(&sums[1], s);
  }
}

// Stage 5: norm rescale with the reference's eps guards.
__global__ __launch_bounds__(256)
void finalize_kernel(const float* __restrict__ z_fb, const float* __restrict__ sums,
                     float* __restrict__ out) {
  const int i = blockIdx.x * 256 + threadIdx.x;
  if (i >= PHASE) return;
  float target  = sqrtf(sums[0]);
  float current = sqrtf(sums[1]);
  float zfb = z_fb[i];
  float v = (current > NORM_EPS) ? zfb * (target / fmaxf(current, NORM_EPS)) : zfb;
  if (!(target > NORM_EPS)) v = 0.0f;
  out[i] = v;
}

// ---------------------------------------------------------------------------
extern "C" void kernel_launch(void* const* d_in, const int* in_sizes, int n_in,
                              void* d_out, int out_size, void* d_ws, size_t ws_size,
                              hipStream_t stream) {
  const float* x     = (const float*)d_in[0];   // active_sdr [65536]
  const float* vel   = (const float*)d_in[1];   // velocity [2]
  const float* W1    = (const float*)d_in[2];   // [4096, 65536]
  const float* b1    = (const float*)d_in[3];   // [4096]
  const float* W2    = (const float*)d_in[4];   // [4096, 4096]
  const float* b2    = (const float*)d_in[5];   // [4096]
  const float* vel_W = (const float*)d_in[6];   // [2048, 2]
  const float* W_tc  = (const float*)d_in[7];   // [4096, 4096]
  const float* W_ht  = (const float*)d_in[8];   // [4096, 4096]
  const float* g_nrt = (const float*)d_in[9];   // [4096]
  const float* h_tc  = (const float*)d_in[10];  // [4096]
  float* out = (float*)d_out;

  float* ws     = (float*)d_ws;
  float* h      = ws;                 // [4096]
  float* z_ego  = ws + 4096;          // [4096]
  float* z_flat = ws + 8192;          // [4096]
  float* z_fb   = ws + 12288;         // [4096]
  float* sums   = ws + 16384;         // [2]: sumsq(z_ego), sumsq(z_fb)

  // 1) h = relu(W1 @ x + b1)            -- 1 GB stream, the dominant stage
  gemv_relu_kernel<<<HIDDEN, 256, 0, stream>>>(W1, x, b1, h);
  // 2) z_ego = W2 @ h + b2              -- WMMA, 256 waves / 32 blocks
  gemv_wmma_bias_kernel<<<PHASE / 16 / 8, 256, 0, stream>>>(W2, h, b2, z_ego, HIDDEN);
  // 3) rotation + ||z_ego||^2
  init_sums_kernel<<<1, 32, 0, stream>>>(sums);
  rotate_kernel<<<N_MOD / 256, 256, 0, stream>>>(z_ego, vel, vel_W, z_flat, sums);
  // 4) feedback path (two WMMA GEMVs fused) + ||z_fb||^2
  gemv_wmma_fb_kernel<<<PHASE / 16 / 8, 256, 0, stream>>>(W_tc, z_flat, W_ht, h_tc,
                                                          g_nrt, z_fb, sums, PHASE);
  // 5) norm rescale
  finalize_kernel<<<PHASE / 256, 256, 0, stream>>>(z_fb, sums, out);
}